// DFBNet_31662498906351
// MI455X (gfx1250) — compile-verified
//
#include <hip/hip_runtime.h>
#include <math.h>

#define BATCH 8
#define CH    512
#define NPIX  1024   // 32*32
#define TOPK  12

typedef float v2f __attribute__((ext_vector_type(2)));
typedef float v8f __attribute__((ext_vector_type(8)));

// ---------------------------------------------------------------- K0: mask counts
__global__ void k_maskcnt(const int* __restrict__ mask, float* cntF, float* cntB) {
    int b = blockIdx.x, t = threadIdx.x;
    int cf = 0, cb = 0;
    for (int n = t; n < NPIX; n += 256) {
        int m = mask[b * NPIX + n];
        cf += (m == 1);
        cb += (m == 0);
    }
    __shared__ int s0[256], s1[256];
    s0[t] = cf; s1[t] = cb; __syncthreads();
    for (int st = 128; st > 0; st >>= 1) {
        if (t < st) { s0[t] += s0[t + st]; s1[t] += s1[t + st]; }
        __syncthreads();
    }
    if (t == 0) { cntF[b] = (float)s0[0]; cntB[b] = (float)s1[0]; }
}

// ---------------------------------------------------------------- K1: masked avg pool protos
__global__ void k_protos(const float* __restrict__ sfeat, const int* __restrict__ mask,
                         const float* cntF, const float* cntB,
                         float* FP, float* BPp) {
    int c = blockIdx.x, b = blockIdx.y, t = threadIdx.x;
    const float* base = sfeat + ((size_t)(b * CH + c)) * NPIX;
    const int* mb = mask + b * NPIX;
    float sF = 0.f, sB = 0.f;
    for (int n = t; n < NPIX; n += 256) {
        float v = base[n];
        int m = mb[n];
        sF += (m == 1) ? v : 0.f;
        sB += (m == 0) ? v : 0.f;
    }
    __shared__ float r0[256], r1[256];
    r0[t] = sF; r1[t] = sB; __syncthreads();
    for (int st = 128; st > 0; st >>= 1) {
        if (t < st) { r0[t] += r0[t + st]; r1[t] += r1[t + st]; }
        __syncthreads();
    }
    if (t == 0) {
        FP[b * CH + c]  = r0[0] / (cntF[b] + 1e-5f);
        BPp[b * CH + c] = r1[0] / (cntB[b] + 1e-5f);
    }
}

// ---------------------------------------------------------------- norm of a (B,C) vector set
__global__ void k_vecnorm(const float* __restrict__ src, float* dst) {
    int b = blockIdx.x, t = threadIdx.x;
    float s = 0.f;
    for (int c = t; c < CH; c += 256) { float v = src[b * CH + c]; s += v * v; }
    __shared__ float r0[256];
    r0[t] = s; __syncthreads();
    for (int st = 128; st > 0; st >>= 1) {
        if (t < st) r0[t] += r0[t + st];
        __syncthreads();
    }
    if (t == 0) dst[b] = sqrtf(r0[0]);
}

// ---------------------------------------------------------------- K2: pred softmax + rnorm
__global__ void k_pred(const float* __restrict__ q, const float* __restrict__ FP,
                       const float* __restrict__ BPp, const float* __restrict__ nFP,
                       const float* __restrict__ nBP,
                       float* predF, float* predB, float* rnorm) {
    int tid = blockIdx.x * 256 + threadIdx.x;
    int b = tid >> 10, n = tid & (NPIX - 1);
    const float* qb = q + ((size_t)b * CH) * NPIX + n;
    const float* fp = FP + b * CH;
    const float* bp = BPp + b * CH;
    float df = 0.f, db = 0.f, qq = 0.f;
    for (int c = 0; c < CH; ++c) {
        float v = qb[(size_t)c * NPIX];
        df += v * fp[c];
        db += v * bp[c];
        qq += v * v;
    }
    float na = sqrtf(qq);
    float sf = 10.f * df / fmaxf(na * nFP[b], 1e-8f);
    float sb = 10.f * db / fmaxf(na * nBP[b], 1e-8f);
    float mx = fmaxf(sf, sb);
    float ef = expf(sf - mx), eb = expf(sb - mx);
    float inv = 1.f / (ef + eb);
    predF[tid] = ef * inv;
    predB[tid] = eb * inv;
    rnorm[tid] = 1.f / na;
}

// ---------------------------------------------------------------- K3: weight select (thres | top-12)
__global__ void k_select(const float* __restrict__ predF, const float* __restrict__ predB,
                         float* wf, float* wb, float* wfs, float* wbs) {
    int b = blockIdx.x, which = blockIdx.y, t = threadIdx.x;
    const float* pred = which ? (predB + b * NPIX) : (predF + b * NPIX);
    float thres = which ? 0.6f : 0.7f;
    float* w = which ? (wb + b * NPIX) : (wf + b * NPIX);
    float* wsum = which ? wbs : wfs;

    float v[4];
    int cnt = 0;
    for (int k = 0; k < 4; ++k) {
        v[k] = pred[k * 256 + t];
        cnt += (v[k] > thres);
    }
    __shared__ int sc[256];
    sc[t] = cnt; __syncthreads();
    for (int st = 128; st > 0; st >>= 1) {
        if (t < st) sc[t] += sc[t + st];
        __syncthreads();
    }
    __shared__ int total;
    if (t == 0) total = sc[0];
    __syncthreads();

    if (total > 0) {
        for (int k = 0; k < 4; ++k) w[k * 256 + t] = (v[k] > thres) ? 1.f : 0.f;
        if (t == 0) wsum[b] = (float)total;
    } else {
        float wm[4] = {0.f, 0.f, 0.f, 0.f};
        __shared__ float sv[256];
        __shared__ int si[256];
        __shared__ int win;
        for (int iter = 0; iter < TOPK; ++iter) {
            float bv = -1e30f; int bi = 1 << 30;
            for (int k = 0; k < 4; ++k) {
                int n = k * 256 + t;
                if (v[k] > bv) { bv = v[k]; bi = n; }
            }
            sv[t] = bv; si[t] = bi; __syncthreads();
            for (int st = 128; st > 0; st >>= 1) {
                if (t < st) {
                    if (sv[t + st] > sv[t] || (sv[t + st] == sv[t] && si[t + st] < si[t])) {
                        sv[t] = sv[t + st]; si[t] = si[t + st];
                    }
                }
                __syncthreads();
            }
            if (t == 0) win = si[0];
            __syncthreads();
            for (int k = 0; k < 4; ++k) {
                int n = k * 256 + t;
                if (n == win) { v[k] = -2.f; wm[k] = 1.f; }
            }
            __syncthreads();
        }
        for (int k = 0; k < 4; ++k) w[k * 256 + t] = wm[k];
        if (t == 0) wsum[b] = (float)TOPK;
    }
}

// ---------------------------------------------------------------- K4: SSP protos + FP1 fuse
__global__ void k_protos2(const float* __restrict__ q, const float* __restrict__ wf,
                          const float* __restrict__ wb, const float* wfs, const float* wbs,
                          const float* __restrict__ FP, float* FP1, float* bgp) {
    int c = blockIdx.x, b = blockIdx.y, t = threadIdx.x;
    const float* qb = q + ((size_t)(b * CH + c)) * NPIX;
    const float* wfb = wf + b * NPIX;
    const float* wbb = wb + b * NPIX;
    float sF = 0.f, sB = 0.f;
    for (int n = t; n < NPIX; n += 256) {
        float v = qb[n];
        sF += v * wfb[n];
        sB += v * wbb[n];
    }
    __shared__ float r0[256], r1[256];
    r0[t] = sF; r1[t] = sB; __syncthreads();
    for (int st = 128; st > 0; st >>= 1) {
        if (t < st) { r0[t] += r0[t + st]; r1[t] += r1[t + st]; }
        __syncthreads();
    }
    if (t == 0) {
        FP1[b * CH + c] = 0.5f * FP[b * CH + c] + 0.5f * (r0[0] / wfs[b]);
        bgp[b * CH + c] = r1[0] / wbs[b];
    }
}

// ---------------------------------------------------------------- K4b: transpose q -> qT (N,C)
__global__ void k_transpose(const float* __restrict__ q, float* __restrict__ qT) {
    __shared__ float tile[32][33];
    int b = blockIdx.z;
    int n0 = blockIdx.x * 32, c0 = blockIdx.y * 32;
    int tx = threadIdx.x, ty = threadIdx.y;   // 32 x 8
    const float* qb = q + (size_t)b * CH * NPIX;
    float* qTb = qT + (size_t)b * NPIX * CH;
    for (int r = 0; r < 32; r += 8)
        tile[ty + r][tx] = qb[(size_t)(c0 + ty + r) * NPIX + n0 + tx];
    __syncthreads();
    for (int r = 0; r < 32; r += 8)
        qTb[(size_t)(n0 + ty + r) * CH + c0 + tx] = tile[tx][ty + r];
}

// ---------------------------------------------------------------- K5: fused WMMA attention
//  Per block: one 16-row tile i of one batch.
//  GEMM1 (WMMA f32 16x16x4):  G[i,j] = sum_c q[c,i] q[c,j]   (K = C = 512)
//  P[i,j] = wb[j] * exp(2 * rnorm[i] * rnorm[j] * G[i,j])     (no max needed: sim in [-2,2])
//  GEMM2 (WMMA f32 16x16x4):  num[c,i] = sum_j q[c,j] P[i,j]  (K = N, chunked by 256)
//  BP1[c,i] = 0.3*bg_proto[c] + 0.7 * num[c,i] / denom[i]
#define PSTR 258   // P_lds row stride in floats (bank-conflict-free)
__global__ __launch_bounds__(256) void k_attn(const float* __restrict__ q,
                                              const float* __restrict__ qT,
                                              const float* __restrict__ rnorm,
                                              const float* __restrict__ wbw,
                                              const float* __restrict__ bgp,
                                              float* __restrict__ BP1) {
    int b = blockIdx.y;
    int i0 = blockIdx.x * 16;
    int t = threadIdx.x;
    int wave = t >> 5;
    int lane = t & 31;
    int m = lane & 15;     // fragment "row/col within 16"
    int kh = lane >> 4;    // half-wave selector

    __shared__ float A_lds[CH * 16];       // raw q[c][i0+m], layout [c][m]
    __shared__ float P_lds[16 * PSTR];     // P chunk [i][j'] (j' in [0,256))
    __shared__ float rn2_i[16];            // 2 * rnorm[i0 + i]
    __shared__ float denom[16];

    const float* qb  = q  + (size_t)b * CH * NPIX;
    const float* qTb = qT + (size_t)b * NPIX * CH;

    // stage A panel with 128-bit transfers (16 floats per c-row, 4 per thread)
    for (int idx4 = t * 4; idx4 < CH * 16; idx4 += 256 * 4) {
        int c = idx4 >> 4, mm = idx4 & 15;
        const float4 v = *(const float4*)&qb[(size_t)c * NPIX + i0 + mm];
        *(float4*)&A_lds[idx4] = v;
    }
    if (t < 16) { rn2_i[t] = 2.f * rnorm[b * NPIX + i0 + t]; denom[t] = 0.f; }
    __syncthreads();

    v8f oacc[4];
    for (int s = 0; s < 4; ++s) oacc[s] = v8f{0.f, 0.f, 0.f, 0.f, 0.f, 0.f, 0.f, 0.f};

    for (int jc = 0; jc < 4; ++jc) {
        int jbase = jc * 256;

        // ---- GEMM1: this wave computes its 2 j-tiles together (shared A frag, 2 indep chains)
        {
            int j0a = jbase + (wave * 32) + m;       // tile 0 column for this lane
            int j0b = j0a + 16;                      // tile 1 column
            v8f acc0 = v8f{0.f, 0.f, 0.f, 0.f, 0.f, 0.f, 0.f, 0.f};
            v8f acc1 = v8f{0.f, 0.f, 0.f, 0.f, 0.f, 0.f, 0.f, 0.f};
            for (int k0 = 0; k0 < CH; k0 += 4) {
                v2f a, b0, b1;
                a.x = A_lds[(k0 + 2 * kh) * 16 + m];
                a.y = A_lds[(k0 + 2 * kh + 1) * 16 + m];
                size_t r0 = (size_t)(k0 + kh) * NPIX;
                size_t r1 = (size_t)(k0 + 2 + kh) * NPIX;
                b0.x = qb[r0 + j0a];
                b0.y = qb[r1 + j0a];
                b1.x = qb[r0 + j0b];
                b1.y = qb[r1 + j0b];
                acc0 = __builtin_amdgcn_wmma_f32_16x16x4_f32(
                        false, a, false, b0, (short)0, acc0, false, false);
                acc1 = __builtin_amdgcn_wmma_f32_16x16x4_f32(
                        false, a, false, b1, (short)0, acc1, false, false);
            }
            // D layout: lane holds column j; rows i = r + 8*kh
            float rja  = rnorm[b * NPIX + j0a];
            float rjb  = rnorm[b * NPIX + j0b];
            float wbja = wbw[b * NPIX + j0a];
            float wbjb = wbw[b * NPIX + j0b];
            int jpa = j0a - jbase;
            int jpb = j0b - jbase;
#pragma unroll
            for (int r = 0; r < 8; ++r) {
                int i = r + 8 * kh;
                float s2 = rn2_i[i];
                P_lds[i * PSTR + jpa] = wbja * expf(acc0[r] * rja * s2);
                P_lds[i * PSTR + jpb] = wbjb * expf(acc1[r] * rjb * s2);
            }
        }
        __syncthreads();

        // ---- denominator partial sums for this chunk
        {
            int i = t >> 4, seg = t & 15;
            float p = 0.f;
            for (int u = 0; u < 16; ++u) p += P_lds[i * PSTR + seg * 16 + u];
            atomicAdd(&denom[i], p);
        }

        // ---- GEMM2: B fragment (from P_lds) shared across the wave's 4 c-tiles
        {
            int cb = wave * 64 + m;                  // c of lane for c-tile s: cb + 16*s
            for (int k0 = 0; k0 < 256; k0 += 4) {
                v2f bb;
                bb.x = P_lds[m * PSTR + k0 + kh];
                bb.y = P_lds[m * PSTR + k0 + 2 + kh];
                int j = jbase + k0 + 2 * kh;
                const float* pa0 = &qTb[(size_t)j * CH + cb];
                const float* pa1 = pa0 + CH;         // j + 1 row
#pragma unroll
                for (int s = 0; s < 4; ++s) {
                    v2f a;
                    a.x = pa0[16 * s];
                    a.y = pa1[16 * s];
                    oacc[s] = __builtin_amdgcn_wmma_f32_16x16x4_f32(
                                false, a, false, bb, (short)0, oacc[s], false, false);
                }
            }
        }
        __syncthreads();   // protect P_lds reuse + make denom adds visible
    }

    // ---- finalize: lane holds column i = i0+m; rows c = c0 + r + 8*kh
    float invd = 1.f / denom[m];
    for (int s = 0; s < 4; ++s) {
        int c0 = (wave * 4 + s) * 16;
#pragma unroll
        for (int r = 0; r < 8; ++r) {
            int c = c0 + r + 8 * kh;
            float v = oacc[s][r] * invd;
            BP1[((size_t)b * CH + c) * NPIX + i0 + m] = 0.3f * bgp[b * CH + c] + 0.7f * v;
        }
    }
}

// ---------------------------------------------------------------- K6: final similarity
__global__ void k_out(const float* __restrict__ q, const float* __restrict__ FP1,
                      const float* __restrict__ nFP1, const float* __restrict__ BP1,
                      const float* __restrict__ rnorm, float* __restrict__ out) {
    int tid = blockIdx.x * 256 + threadIdx.x;
    int b = tid >> 10, n = tid & (NPIX - 1);
    const float* qb = q + (size_t)b * CH * NPIX + n;
    const float* bb = BP1 + (size_t)b * CH * NPIX + n;
    const float* fp = FP1 + b * CH;
    float df = 0.f, db = 0.f, nb = 0.f;
    for (int c = 0; c < CH; ++c) {
        float v = qb[(size_t)c * NPIX];
        float w = bb[(size_t)c * NPIX];
        df += v * fp[c];
        db += v * w;
        nb += w * w;
    }
    float na = 1.f / rnorm[tid];
    float sb = 10.f * db / fmaxf(na * sqrtf(nb), 1e-8f);
    float sf = 10.f * df / fmaxf(na * nFP1[b], 1e-8f);
    out[((size_t)b * 2) * NPIX + n]     = sb;   // channel 0: bg
    out[((size_t)b * 2 + 1) * NPIX + n] = sf;   // channel 1: fg
}

// ---------------------------------------------------------------- launcher
extern "C" void kernel_launch(void* const* d_in, const int* in_sizes, int n_in,
                              void* d_out, int out_size, void* d_ws, size_t ws_size,
                              hipStream_t stream) {
    const float* q    = (const float*)d_in[0];  // feature_q   (B,C,H,W)
    const float* sf   = (const float*)d_in[1];  // support_feat(B,C,H,W)
    const int*   mask = (const int*)d_in[2];    // support_mask(B,H,W)
    float* out = (float*)d_out;

    float* w = (float*)d_ws;
    size_t off = 0;
    auto alloc = [&](size_t nelem) { float* p = w + off; off += nelem; return p; };

    float* cntF  = alloc(BATCH);
    float* cntB  = alloc(BATCH);
    float* FP    = alloc((size_t)BATCH * CH);
    float* BPp   = alloc((size_t)BATCH * CH);
    float* nFP   = alloc(BATCH);
    float* nBP   = alloc(BATCH);
    float* predF = alloc((size_t)BATCH * NPIX);
    float* predB = alloc((size_t)BATCH * NPIX);
    float* rnorm = alloc((size_t)BATCH * NPIX);
    float* wf    = alloc((size_t)BATCH * NPIX);
    float* wb    = alloc((size_t)BATCH * NPIX);
    float* wfs   = alloc(BATCH);
    float* wbs   = alloc(BATCH);
    float* FP1   = alloc((size_t)BATCH * CH);
    float* bgp   = alloc((size_t)BATCH * CH);
    float* nFP1  = alloc(BATCH);
    float* qT    = alloc((size_t)BATCH * NPIX * CH);   // 16 MB
    float* BP1   = alloc((size_t)BATCH * CH * NPIX);   // 16 MB

    k_maskcnt<<<BATCH, 256, 0, stream>>>(mask, cntF, cntB);
    k_protos<<<dim3(CH, BATCH), 256, 0, stream>>>(sf, mask, cntF, cntB, FP, BPp);
    k_vecnorm<<<BATCH, 256, 0, stream>>>(FP, nFP);
    k_vecnorm<<<BATCH, 256, 0, stream>>>(BPp, nBP);
    k_pred<<<(BATCH * NPIX) / 256, 256, 0, stream>>>(q, FP, BPp, nFP, nBP, predF, predB, rnorm);
    k_select<<<dim3(BATCH, 2), 256, 0, stream>>>(predF, predB, wf, wb, wfs, wbs);
    k_protos2<<<dim3(CH, BATCH), 256, 0, stream>>>(q, wf, wb, wfs, wbs, FP, FP1, bgp);
    k_vecnorm<<<BATCH, 256, 0, stream>>>(FP1, nFP1);
    k_transpose<<<dim3(NPIX / 32, CH / 32, BATCH), dim3(32, 8), 0, stream>>>(q, qT);
    k_attn<<<dim3(NPIX / 16, BATCH), 256, 0, stream>>>(q, qT, rnorm, wb, bgp, BP1);
    k_out<<<(BATCH * NPIX) / 256, 256, 0, stream>>>(q, FP1, nFP1, BP1, rnorm, out);
}